// Reservoir_35399120453974
// MI455X (gfx1250) — compile-verified
//
#include <hip/hip_runtime.h>
#include <hip/hip_bf16.h>

// ESN forward for MI455X (gfx1250), wave32, WMMA f32 16x16x4.
//
// x:      [B=32, S=256, N=64, F=32]  f32
// w_ih0:  [H=128, F=32]   b_ih0: [128]   w_hh0: [128,128]
// w_ih1:  [128, 128]      b_ih1: [128]   w_hh1: [128,128]
// out:    [B, S, N, 2*H] f32   (layer0 cols 0..127, layer1 cols 128..255)
//
// Grid: 128 workgroups x 256 threads. Each workgroup owns 16 BN rows and
// runs the full 256-step recurrence with h0/h1 resident in LDS/registers.
// Weights are held as WMMA B-fragments in registers (loaded once).

typedef __attribute__((ext_vector_type(2))) float v2f;
typedef __attribute__((ext_vector_type(8))) float v8f;

#define ESN_H 128
#define ESN_F 32
#define ESN_S 256
#define ESN_N 64
#define ESN_ALPHA 0.9f
#define HPAD 132   // padded LDS row stride (floats); 132%64=4 -> no bank collapse

// Native CDNA5 V_TANH_F32 (TRANS op, co-executes with the WMMA pipe) when
// available; fall back to libm otherwise.
#if __has_builtin(__builtin_amdgcn_tanhf)
#define ESN_TANH(x) __builtin_amdgcn_tanhf(x)
#elif __has_builtin(__builtin_amdgcn_tanh_f32)
#define ESN_TANH(x) __builtin_amdgcn_tanh_f32(x)
#else
#define ESN_TANH(x) tanhf(x)
#endif

static __device__ __forceinline__ v8f wmma_f32_k4(v2f a, v2f b, v8f c) {
  // D = A(16x4) * B(4x16) + C(16x16), all fp32.
  return __builtin_amdgcn_wmma_f32_16x16x4_f32(
      /*neg_a=*/false, a, /*neg_b=*/false, b,
      /*c_mod=*/(short)0, c, /*reuse_a=*/false, /*reuse_b=*/false);
}

__global__ __launch_bounds__(256)
void esn2_wmma_kernel(const float* __restrict__ x,
                      const float* __restrict__ wih0,
                      const float* __restrict__ bih0,
                      const float* __restrict__ whh0,
                      const float* __restrict__ wih1,
                      const float* __restrict__ bih1,
                      const float* __restrict__ whh1,
                      float* __restrict__ out) {
  __shared__ float h0s[16 * HPAD];
  __shared__ float h1s[16 * HPAD];

  const int lane = threadIdx.x & 31;
  const int wave = threadIdx.x >> 5;     // 0..7 -> output column tile
  const int n0   = wave << 4;            // first output column of this wave
  const int nn   = lane & 15;            // column within tile / A-row m
  const int hi   = lane >> 4;            // half-wave select (K split for A/B)
  const int m    = nn;                   // A-matrix row for this lane

  const int row0  = blockIdx.x << 4;     // BN row base (BN = b*64 + n)
  const int bidx  = row0 >> 6;
  const int nidx0 = row0 & 63;

  // ---- preload weight B-fragments into registers (constant across steps) ----
  // B[k][n] = w[n][k]; lane holds column n = n0+nn, fragment j covers k = 4j..4j+3,
  // this lane supplies {w[n][4j+2hi], w[n][4j+2hi+1]}.
  v2f Bih0[ESN_F / 4], Bhh0[ESN_H / 4], Bih1[ESN_H / 4], Bhh1[ESN_H / 4];
  {
    const int n = n0 + nn;
#pragma unroll
    for (int j = 0; j < ESN_F / 4; ++j)
      Bih0[j] = *(const v2f*)(wih0 + n * ESN_F + 4 * j + 2 * hi);
#pragma unroll
    for (int j = 0; j < ESN_H / 4; ++j) {
      Bhh0[j] = *(const v2f*)(whh0 + n * ESN_H + 4 * j + 2 * hi);
      Bih1[j] = *(const v2f*)(wih1 + n * ESN_H + 4 * j + 2 * hi);
      Bhh1[j] = *(const v2f*)(whh1 + n * ESN_H + 4 * j + 2 * hi);
    }
  }
  const float bias0 = bih0[n0 + nn];
  const float bias1 = bih1[n0 + nn];

  // ---- zero recurrent state ----
  for (int i = threadIdx.x; i < 16 * HPAD; i += 256) {
    h0s[i] = 0.0f;
    h1s[i] = 0.0f;
  }
  v8f h0r = {};  // this wave's 16x16 chunk of h0 (C/D layout)
  v8f h1r = {};
  __syncthreads();

  const long xbase = ((long)bidx * ESN_S * ESN_N + nidx0) * ESN_F;
  const long obase = ((long)bidx * ESN_S * ESN_N + nidx0) * (2 * ESN_H);

  for (int s = 0; s < ESN_S; ++s) {
    const float* xs = x + xbase + (long)s * (ESN_N * ESN_F);
    if (s + 1 < ESN_S)
      __builtin_prefetch(xs + ESN_N * ESN_F + m * ESN_F, 0, 3);

    // ---- layer 0: acc0 = b0 + x_s @ Wih0^T + h0_old @ Whh0^T ----
    v8f acc0;
#pragma unroll
    for (int r = 0; r < 8; ++r) acc0[r] = bias0;
#pragma unroll
    for (int j = 0; j < ESN_F / 4; ++j) {
      v2f a = *(const v2f*)(xs + m * ESN_F + 4 * j + 2 * hi);
      acc0 = wmma_f32_k4(a, Bih0[j], acc0);
    }
#pragma unroll
    for (int j = 0; j < ESN_H / 4; ++j) {
      v2f a = *(const v2f*)(h0s + m * HPAD + 4 * j + 2 * hi);
      acc0 = wmma_f32_k4(a, Bhh0[j], acc0);
    }

    // ---- layer 1 partial: acc1 = b1 + h1_old @ Whh1^T (uses old h1 LDS) ----
    v8f acc1;
#pragma unroll
    for (int r = 0; r < 8; ++r) acc1[r] = bias1;
#pragma unroll
    for (int j = 0; j < ESN_H / 4; ++j) {
      v2f a = *(const v2f*)(h1s + m * HPAD + 4 * j + 2 * hi);
      acc1 = wmma_f32_k4(a, Bhh1[j], acc1);
    }

    // ---- h0_new = (1-a)*h0_old + a*tanh(acc0) ----
    v8f h0n;
#pragma unroll
    for (int r = 0; r < 8; ++r)
      h0n[r] = (1.0f - ESN_ALPHA) * h0r[r] + ESN_ALPHA * ESN_TANH(acc0[r]);

    __syncthreads();  // all reads of old h0/h1 LDS complete
#pragma unroll
    for (int r = 0; r < 8; ++r)
      h0s[(r + 8 * hi) * HPAD + n0 + nn] = h0n[r];
    __syncthreads();  // h0_new visible to all waves

    // ---- layer 1 finish: acc1 += h0_new @ Wih1^T ----
#pragma unroll
    for (int j = 0; j < ESN_H / 4; ++j) {
      v2f a = *(const v2f*)(h0s + m * HPAD + 4 * j + 2 * hi);
      acc1 = wmma_f32_k4(a, Bih1[j], acc1);
    }

    v8f h1n;
#pragma unroll
    for (int r = 0; r < 8; ++r)
      h1n[r] = (1.0f - ESN_ALPHA) * h1r[r] + ESN_ALPHA * ESN_TANH(acc1[r]);
#pragma unroll
    for (int r = 0; r < 8; ++r)
      h1s[(r + 8 * hi) * HPAD + n0 + nn] = h1n[r];

    // ---- write out[b, s, n, 0:128]=h0, [128:256]=h1 ----
    float* o = out + obase + (long)s * (ESN_N * 2 * ESN_H);
#pragma unroll
    for (int r = 0; r < 8; ++r) {
      const int mm = r + 8 * hi;
      o[mm * (2 * ESN_H) + n0 + nn]         = h0n[r];
      o[mm * (2 * ESN_H) + ESN_H + n0 + nn] = h1n[r];
    }

    h0r = h0n;
    h1r = h1n;
    __syncthreads();  // h1_new stores complete before next step's reads
  }
}

extern "C" void kernel_launch(void* const* d_in, const int* in_sizes, int n_in,
                              void* d_out, int out_size, void* d_ws, size_t ws_size,
                              hipStream_t stream) {
  const float* x    = (const float*)d_in[0];
  const float* wih0 = (const float*)d_in[1];
  const float* bih0 = (const float*)d_in[2];
  const float* whh0 = (const float*)d_in[3];
  const float* wih1 = (const float*)d_in[4];
  const float* bih1 = (const float*)d_in[5];
  const float* whh1 = (const float*)d_in[6];
  // d_in[7] = layer_num (always 2 for this problem)
  float* out = (float*)d_out;

  // BN = 32*64 = 2048 rows; 16 rows per workgroup -> 128 workgroups.
  dim3 grid(128), block(256);
  esn2_wmma_kernel<<<grid, block, 0, stream>>>(x, wih0, bih0, whh0,
                                               wih1, bih1, whh1, out);
}